// Semantic_Mapping_71416716197979
// MI455X (gfx1250) — compile-verified
//
#include <hip/hip_runtime.h>
#include <hip/hip_bf16.h>
#include <cstdint>

// ---------------- problem constants ----------------
#define BSZ        4
#define NSEM       24
#define CCH        28          // 4 + NUM_SEM
#define FEATC      25          // 1 + NUM_SEM
#define HS         240
#define WS_        320
#define VRN        100
#define NZ         80          // MAX_H - MIN_H
#define MC         480         // MAP_CELLS
#define LOOK       96          // VR - 4
#define MINZ       11
#define MAXZ       49          // exclusive
#define XC_        159.5f
#define ZC_        119.5f
#define FOCAL_     277.12812921102035f
#define AGH        155.0f
#define SHX        250.0f
#define D2R        0.017453292519943295f
#define R2D_       57.29577951308232f

// workspace layout (float offsets)
#define PA_OFF     0                       // projAgent: 4*25*100*100 = 1,000,000
#define PALL_OFF   1000000                 // projAll:   4*100*100    =    40,000
#define AV_OFF     1040000                 // av tiles:  4*28*96*100  = 1,075,200
#define CST_OFF    2115200                 // consts:    4*4 = 16
#define GRID_OFF   2115216                 // voxel chunks start here
#define FIXED_BYTES (GRID_OFF * 4ull)
#define GRID_FLOATS 800000ull              // 100*100*80 per (b,ch) pair
#define NPAIRS     100                     // 4 * 25

// output layout (float offsets)
#define OUT_FPM    0                       // 4*1*100*100 = 40,000
#define OUT_MAP    40000                   // 4*28*480*480 = 25,804,800
#define OUT_P1     25844800
#define OUT_P2     25844812

#if defined(__has_builtin)
#  if __has_builtin(__builtin_amdgcn_global_load_async_to_lds_b128)
#    define HAVE_ASYNC_LDS 1
#  endif
#endif

typedef int v4i_t __attribute__((ext_vector_type(4)));

__device__ __forceinline__ float clamp01(float v) {
    return fminf(fmaxf(v, 0.f), 1.f);
}

// ---------------- K0: pose composition + per-batch sample constants ----------------
__global__ void k_pose(const float* __restrict__ pose_obs,
                       const float* __restrict__ poses_last,
                       float* __restrict__ out,
                       float* __restrict__ consts) {
    int b = threadIdx.x;
    if (b >= BSZ) return;
    float th0 = poses_last[b * 3 + 2] / R2D_;
    float s = sinf(th0), c = cosf(th0);
    float nx = poses_last[b * 3 + 0] + pose_obs[b * 3 + 0] * c - pose_obs[b * 3 + 1] * s;
    float ny = poses_last[b * 3 + 1] + pose_obs[b * 3 + 0] * s + pose_obs[b * 3 + 1] * c;
    float no = poses_last[b * 3 + 2] + pose_obs[b * 3 + 2] * R2D_;
    no = fmodf(no - 180.f, 360.f) + 180.f;
    no = fmodf(no + 180.f, 360.f) - 180.f;
    out[OUT_P1 + b * 3 + 0] = nx;  out[OUT_P1 + b * 3 + 1] = ny;  out[OUT_P1 + b * 3 + 2] = no;
    out[OUT_P2 + b * 3 + 0] = nx;  out[OUT_P2 + b * 3 + 1] = ny;  out[OUT_P2 + b * 3 + 2] = no;
    // translation grid offsets (pixels) and rotation angle
    float sx = -(nx * 20.f - 240.f) / 240.f;    // nx*100/RES = nx*20 ; half = 240
    float sy = -(ny * 20.f - 240.f) / 240.f;
    float t  = (90.f - no) * D2R;
    consts[b * 4 + 0] = cosf(t);
    consts[b * 4 + 1] = sinf(t);
    consts[b * 4 + 2] = sx * 239.5f;            // (MC-1)/2
    consts[b * 4 + 3] = sy * 239.5f;
}

// ---------------- K1: trilinear splat (scatter-add) into voxel chunk ----------------
__global__ void k_splat(const float* __restrict__ obs,
                        const float* __restrict__ view_angles,
                        float* __restrict__ grid,   // chunk base
                        int p0, int p1) {
    int t = blockIdx.x * blockDim.x + threadIdx.x;
    const int PIX = HS * WS_;
    if (t >= BSZ * PIX) return;
    int b   = t / PIX;
    int pix = t - b * PIX;
    int r   = pix / WS_;
    int cx  = pix - r * WS_;

    int cs = p0 - b * FEATC; if (cs < 0) cs = 0;
    int ce = p1 - b * FEATC; if (ce > FEATC) ce = FEATC;
    if (cs >= ce) return;

    float d = obs[(((size_t)b * CCH + 3) * HS + r) * WS_ + cx];
    if (d < 0.f) d = 10000.f;
    float X  = ((float)cx - XC_) * d / FOCAL_;
    float Zc = ((float)(HS - 1 - r) - ZC_) * d / FOCAL_;
    float a  = view_angles[b] * D2R;
    float ca = cosf(a), sa = sinf(a);
    float Yw = ca * d - sa * Zc;
    float Zw = sa * d + ca * Zc + AGH;
    float Xw = X + SHX;

    float px = Xw * 0.2f;            // /RES
    float py = Yw * 0.2f;
    float pz = Zw * 0.2f + 8.f;      // /Z_RES - MIN_H

    float flx = floorf(px), fly = floorf(py), flz = floorf(pz);
    int   ix[2]; float wx[2];
    int   iy[2]; float wy[2];
    int   iz[2]; float wz[2];
#pragma unroll
    for (int o = 0; o < 2; ++o) {
        float p, w; int idx;
        p = flx + (float)o; w = (p > 0.f && p < (float)VRN) ? (1.f - fabsf(px - p)) : 0.f;
        idx = (p > 0.f && p < (float)VRN) ? (int)p : 0; ix[o] = idx; wx[o] = w;
        p = fly + (float)o; w = (p > 0.f && p < (float)VRN) ? (1.f - fabsf(py - p)) : 0.f;
        idx = (p > 0.f && p < (float)VRN) ? (int)p : 0; iy[o] = idx; wy[o] = w;
        p = flz + (float)o; w = (p > 0.f && p < (float)NZ) ? (1.f - fabsf(pz - p)) : 0.f;
        idx = (p > 0.f && p < (float)NZ) ? (int)p : 0; iz[o] = idx; wz[o] = w;
    }

    for (int ch = cs; ch < ce; ++ch) {
        float val = (ch == 0) ? 1.f
                              : obs[(((size_t)b * CCH + 3 + ch) * HS + r) * WS_ + cx];
        float* g = grid + (size_t)(b * FEATC + ch - p0) * GRID_FLOATS;
#pragma unroll
        for (int ox = 0; ox < 2; ++ox)
#pragma unroll
            for (int oy = 0; oy < 2; ++oy)
#pragma unroll
                for (int oz = 0; oz < 2; ++oz) {
                    float w = wx[ox] * wy[oy] * wz[oz];
                    if (w != 0.f)
                        atomicAdd(&g[((size_t)(ix[ox] * VRN + iy[oy])) * NZ + iz[oz]], val * w);
                }
    }
}

// ---------------- K2: per-voxel round (RNE, like jnp.round) + z reductions ----------------
__global__ void k_reduce(const float* __restrict__ grid,
                         float* __restrict__ projAgent,
                         float* __restrict__ projAll,
                         int p0, int pc) {
    int t = blockIdx.x * blockDim.x + threadIdx.x;
    if (t >= pc * VRN * VRN) return;
    int pl  = t / (VRN * VRN);
    int rem = t - pl * (VRN * VRN);
    int j   = rem / VRN;        // X index (contiguous in grid major)
    int i   = rem - j * VRN;    // Y index
    int pair = p0 + pl;
    int b = pair / FEATC, ch = pair - b * FEATC;

    const float* g = grid + (size_t)pl * GRID_FLOATS + (size_t)(j * VRN + i) * NZ;
    float sBand = 0.f, sAll = 0.f;
#pragma unroll 4
    for (int z = 0; z < NZ; ++z) {
        float v = rintf(g[z]);             // round half-to-even
        sAll += v;
        if (z >= MINZ && z < MAXZ) sBand += v;
    }
    // voxels transposed (0,1,3,2,4): proj[b,ch,Y=i,X=j] = sum_z grid[b,ch,X=j,Y=i,z]
    projAgent[(((size_t)b * FEATC + ch) * VRN + i) * VRN + j] = sBand;
    if (ch == 0) projAll[((size_t)b * VRN + i) * VRN + j] = sAll;
}

// ---------------- K3: compact agent_view tiles (clips folded in) ----------------
__global__ void k_av(const float* __restrict__ projAgent,
                     const float* __restrict__ projAll,
                     float* __restrict__ av) {
    int idx = blockIdx.x * blockDim.x + threadIdx.x;
    const int PERB = CCH * LOOK * VRN;
    if (idx >= BSZ * PERB) return;
    int b   = idx / PERB;
    int rem = idx - b * PERB;
    int ch  = rem / (LOOK * VRN);
    int rr  = rem - ch * (LOOK * VRN);
    int i   = rr / VRN, j = rr - (rr / VRN) * VRN;
    float v = 0.f;
    if (ch == 0)      v = clamp01(projAgent[(((size_t)b * FEATC) * VRN + i) * VRN + j]);
    else if (ch == 1) v = clamp01(projAll[((size_t)b * VRN + i) * VRN + j]);
    else if (ch >= 4) v = clamp01(projAgent[(((size_t)b * FEATC + (ch - 3)) * VRN + i) * VRN + j] * 0.2f);
    av[idx] = v;
}

// ---------------- K4: fp_map_pred output ----------------
__global__ void k_fpmap(const float* __restrict__ projAgent, float* __restrict__ out) {
    int idx = blockIdx.x * blockDim.x + threadIdx.x;
    if (idx >= BSZ * VRN * VRN) return;
    int b = idx / (VRN * VRN);
    int rem = idx - b * (VRN * VRN);
    out[OUT_FPM + idx] = clamp01(projAgent[(size_t)b * FEATC * VRN * VRN + rem]);
}

// ---------------- K5: fused rotate -> translate -> max(maps_last, .) ----------------
__device__ __forceinline__ float av_at(const float* tile, int ya, int xa) {
    int il = ya - 240, jl = xa - 190;   // y1 = MC/2 ; x1 = MC/2 - VR/2
    if ((unsigned)il >= (unsigned)LOOK || (unsigned)jl >= (unsigned)VRN) return 0.f;
    return tile[il * VRN + jl];
}

__device__ __forceinline__ float rot_sample(const float* tile, float ct, float st,
                                            int xn, int yn) {
    const float S = 2.f / 479.f;
    float gx = (float)xn * S - 1.f;
    float gy = (float)yn * S - 1.f;
    float ix = (ct * gx - st * gy) * 239.5f + 239.5f;
    float iy = (st * gx + ct * gy) * 239.5f + 239.5f;
    float fx0 = floorf(ix), fy0 = floorf(iy);
    float wx1 = ix - fx0, wy1 = iy - fy0;
    float wx0 = 1.f - wx1, wy0 = 1.f - wy1;
    int x0 = (int)fx0, y0 = (int)fy0;
    float v00 = av_at(tile, y0,     x0);
    float v01 = av_at(tile, y0,     x0 + 1);
    float v10 = av_at(tile, y0 + 1, x0);
    float v11 = av_at(tile, y0 + 1, x0 + 1);
    return v00 * wx0 * wy0 + v01 * wx1 * wy0 + v10 * wx0 * wy1 + v11 * wx1 * wy1;
}

__global__ void __launch_bounds__(256) k_map(const float* __restrict__ av,
                                             const float* __restrict__ maps_last,
                                             const float* __restrict__ consts,
                                             float* __restrict__ out) {
    __shared__ float tile[LOOK * VRN];          // 38.4 KB
    int band = blockIdx.x;                      // 0..29 (16 rows each)
    int z    = blockIdx.y;                      // 0..111 : (b, ch)
    int b    = z / CCH, ch = z - b * CCH;
    int y0   = band * 16;
    int tid  = threadIdx.x;

    const float* ml = maps_last + ((size_t)z * MC + y0) * MC;
    float* om = out + OUT_MAP + ((size_t)z * MC + y0) * MC;

    // prefetch the streaming maps_last band (global_prefetch_b8)
    if (tid < 240) __builtin_prefetch(ml + tid * 32, 0, 0);

    if (ch == 2 || ch == 3) {                   // agent_view channel identically zero
        for (int k = tid; k < 16 * MC; k += 256) om[k] = fmaxf(ml[k], 0.f);
        return;
    }

    const float* src = av + (size_t)z * (LOOK * VRN);
#if defined(HAVE_ASYNC_LDS)
    for (int k = tid; k < (LOOK * VRN) / 4; k += 256) {
        __builtin_amdgcn_global_load_async_to_lds_b128(
            (__attribute__((address_space(1))) v4i_t*)(src + k * 4),
            (__attribute__((address_space(3))) v4i_t*)(&tile[k * 4]),
            0, 0);
    }
#  if __has_builtin(__builtin_amdgcn_s_wait_asynccnt)
    __builtin_amdgcn_s_wait_asynccnt(0);
#  else
    asm volatile("s_wait_asynccnt 0" ::: "memory");
#  endif
    __syncthreads();
#else
    for (int k = tid; k < (LOOK * VRN) / 4; k += 256)
        reinterpret_cast<float4*>(tile)[k] = reinterpret_cast<const float4*>(src)[k];
    __syncthreads();
#endif

    float ct  = consts[b * 4 + 0];
    float st  = consts[b * 4 + 1];
    float dxp = consts[b * 4 + 2];
    float dyp = consts[b * 4 + 3];

    for (int k = tid; k < 16 * MC; k += 256) {
        int y = y0 + k / MC;
        int x = k - (k / MC) * MC;
        float ix = (float)x + dxp;
        float iy = (float)y + dyp;
        float fx0 = floorf(ix), fy0 = floorf(iy);
        float wx1 = ix - fx0, wy1 = iy - fy0;
        int x0 = (int)fx0, yy0 = (int)fy0;
        float acc = 0.f;
#pragma unroll
        for (int dy = 0; dy < 2; ++dy)
#pragma unroll
            for (int dx = 0; dx < 2; ++dx) {
                int xn = x0 + dx, yn = yy0 + dy;
                float w = (dx ? wx1 : 1.f - wx1) * (dy ? wy1 : 1.f - wy1);
                if (xn >= 0 && xn < MC && yn >= 0 && yn < MC)
                    acc += w * rot_sample(tile, ct, st, xn, yn);
            }
        om[k] = fmaxf(ml[k], acc);
    }
}

// ---------------- host launcher ----------------
extern "C" void kernel_launch(void* const* d_in, const int* in_sizes, int n_in,
                              void* d_out, int out_size, void* d_ws, size_t ws_size,
                              hipStream_t stream) {
    (void)in_sizes; (void)n_in; (void)out_size;
    const float* obs         = (const float*)d_in[0];
    const float* pose_obs    = (const float*)d_in[1];
    const float* maps_last   = (const float*)d_in[2];
    const float* poses_last  = (const float*)d_in[3];
    const float* view_angles = (const float*)d_in[4];
    float* out = (float*)d_out;
    float* ws  = (float*)d_ws;

    float* projAgent = ws + PA_OFF;
    float* projAll   = ws + PALL_OFF;
    float* av        = ws + AV_OFF;
    float* consts    = ws + CST_OFF;
    float* gridBase  = ws + GRID_OFF;

    // zero projAgent and projAll (k_reduce then stores every element)
    (void)hipMemsetAsync(projAgent, 0, (size_t)AV_OFF * sizeof(float), stream);

    k_pose<<<1, 32, 0, stream>>>(pose_obs, poses_last, out, consts);

    // chunk (b,ch) pairs through whatever voxel scratch fits
    size_t avail = (ws_size > FIXED_BYTES) ? (ws_size - FIXED_BYTES) : 0;
    int cap = (int)(avail / (GRID_FLOATS * sizeof(float)));
    if (cap < 1) cap = 1;
    if (cap > NPAIRS) cap = NPAIRS;

    const int splatThreads = BSZ * HS * WS_;
    const int splatBlocks  = (splatThreads + 255) / 256;
    for (int p0 = 0; p0 < NPAIRS; p0 += cap) {
        int pc = (NPAIRS - p0 < cap) ? (NPAIRS - p0) : cap;
        (void)hipMemsetAsync(gridBase, 0, (size_t)pc * GRID_FLOATS * sizeof(float), stream);
        k_splat<<<splatBlocks, 256, 0, stream>>>(obs, view_angles, gridBase, p0, p0 + pc);
        int rt = pc * VRN * VRN;
        k_reduce<<<(rt + 255) / 256, 256, 0, stream>>>(gridBase, projAgent, projAll, p0, pc);
    }

    int avTot = BSZ * CCH * LOOK * VRN;
    k_av<<<(avTot + 255) / 256, 256, 0, stream>>>(projAgent, projAll, av);
    k_fpmap<<<(BSZ * VRN * VRN + 255) / 256, 256, 0, stream>>>(projAgent, out);

    dim3 grid(MC / 16, BSZ * CCH, 1);   // 30 x 112
    k_map<<<grid, 256, 0, stream>>>(av, maps_last, consts, out);
}